// QSA_11063835754517
// MI455X (gfx1250) — compile-verified
//
#include <hip/hip_runtime.h>
#include <hip/hip_bf16.h>

#define NW    10
#define DIM   1024
#define HID   64
#define WAVES 4
#define MROWS 2     // 16-row tiles per wave sharing B-fragment loads

typedef __attribute__((ext_vector_type(2))) float v2f;
typedef __attribute__((ext_vector_type(8))) float v8f;

// ---------------------------------------------------------------------------
// Kernel 1: build the 1024x64 column slice of the circuit unitary (real+imag)
// by simulating each basis column in LDS, plus the +-1 Pauli-Z sign matrix.
// One block per column c; 256 threads; 8 KB LDS state.
// ---------------------------------------------------------------------------
__global__ __launch_bounds__(256)
void build_unitary(const float* __restrict__ rx0, const float* __restrict__ ry0,
                   const float* __restrict__ ry1,
                   float* __restrict__ Wr, float* __restrict__ Wi,
                   float* __restrict__ St) {
    __shared__ float re[DIM];
    __shared__ float im[DIM];
    const int c = blockIdx.x;     // column / basis state 0..63
    const int t = threadIdx.x;    // 0..255

    for (int i = t; i < DIM; i += 256) { re[i] = (i == c) ? 1.0f : 0.0f; im[i] = 0.0f; }
    __syncthreads();

    // Layer 1: RX(rx0[j]) then RY(ry0[j]) on qubit j (qubit j = bit 9-j).
    // Combined G = RY*RX.
    for (int j = 0; j < NW; ++j) {
        const int p = NW - 1 - j;
        const int stride = 1 << p;
        const float cx = cosf(0.5f * rx0[j]), sx = sinf(0.5f * rx0[j]);
        const float cy = cosf(0.5f * ry0[j]), sy = sinf(0.5f * ry0[j]);
        const float g00r =  cy * cx, g00i =  sy * sx;
        const float g01r = -sy * cx, g01i = -cy * sx;
        const float g10r =  sy * cx, g10i = -cy * sx;
        const float g11r =  cy * cx, g11i = -sy * sx;
        for (int q = t; q < DIM / 2; q += 256) {
            const int i0 = ((q >> p) << (p + 1)) | (q & (stride - 1));
            const int i1 = i0 | stride;
            const float a0r = re[i0], a0i = im[i0];
            const float a1r = re[i1], a1i = im[i1];
            re[i0] = g00r * a0r - g00i * a0i + g01r * a1r - g01i * a1i;
            im[i0] = g00r * a0i + g00i * a0r + g01r * a1i + g01i * a1r;
            re[i1] = g10r * a0r - g10i * a0i + g11r * a1r - g11i * a1i;
            im[i1] = g10r * a0i + g10i * a0r + g11r * a1i + g11i * a1r;
        }
        __syncthreads();
    }

    // CNOT ring: control qubit j, target qubit (j+1)%NW.
    for (int j = 0; j < NW; ++j) {
        const int cb = 1 << (NW - 1 - j);
        const int tb = 1 << (NW - 1 - ((j + 1) % NW));
        for (int i = t; i < DIM; i += 256) {
            if ((i & cb) && !(i & tb)) {
                const int ip = i | tb;
                const float r0 = re[i], v0 = im[i];
                re[i] = re[ip]; im[i] = im[ip];
                re[ip] = r0;    im[ip] = v0;
            }
        }
        __syncthreads();
    }

    // Layer 3: RY(ry1[j]) on qubit j (real 2x2).
    for (int j = 0; j < NW; ++j) {
        const int p = NW - 1 - j;
        const int stride = 1 << p;
        const float cy = cosf(0.5f * ry1[j]), sy = sinf(0.5f * ry1[j]);
        for (int q = t; q < DIM / 2; q += 256) {
            const int i0 = ((q >> p) << (p + 1)) | (q & (stride - 1));
            const int i1 = i0 | stride;
            const float a0r = re[i0], a0i = im[i0];
            const float a1r = re[i1], a1i = im[i1];
            re[i0] = cy * a0r - sy * a1r;  im[i0] = cy * a0i - sy * a1i;
            re[i1] = sy * a0r + cy * a1r;  im[i1] = sy * a0i + cy * a1i;
        }
        __syncthreads();
    }

    // Write column c of U, row-major [d][c] for packed B-fragment loads later.
    for (int i = t; i < DIM; i += 256) {
        Wr[(size_t)i * HID + c] = re[i];
        Wi[(size_t)i * HID + c] = im[i];
    }
    // Sign matrix St[d][h] = 1 - 2*parity(popcount(d & (h+1))); block c covers 16 d rows.
    for (int e = t; e < 16 * HID; e += 256) {
        const int d = c * 16 + (e / HID);
        const int h = e % HID;
        const int pc = __popc(d & (h + 1));
        St[(size_t)d * HID + h] = 1.0f - 2.0f * (float)(pc & 1);
    }
}

// ---------------------------------------------------------------------------
// Kernel 2: fused  Xn @ [Wr|Wi]^T  -> square-add -> @ St  using fp32 WMMA.
// Each wave owns MROWS x 16 token rows; both row tiles share every B-fragment
// load (Wr/Wi/St), halving L2 traffic and interleaving independent WMMAs.
// f32 16x16x4 fragment layout: A/B lane L holds (row/col = L%16,
// K = {2*(L/16), 2*(L/16)+1});  C/D vgpr r, lane L -> row r+8*(L/16), col L%16.
// ---------------------------------------------------------------------------
__global__ __launch_bounds__(32 * WAVES)
void qsa_fused(const float* __restrict__ X, const float* __restrict__ Wr,
               const float* __restrict__ Wi, const float* __restrict__ St,
               float* __restrict__ Out) {
    __shared__ float lds[WAVES][MROWS][16 * 17];

    const int wave  = threadIdx.x >> 5;
    const int lane  = threadIdx.x & 31;
    const int mrow  = lane & 15;      // row (A) / col (B) within 16
    const int khalf = lane >> 4;      // which K pair
    const int R     = blockIdx.x * (16 * MROWS * WAVES) + wave * (16 * MROWS);

    // --- Load the wave's MROWSx16x64 X tiles as A-fragments + row norms ---
    v2f a[MROWS][16];
#pragma unroll
    for (int mt = 0; mt < MROWS; ++mt) {
        const float* xrow = X + (size_t)(R + mt * 16 + mrow) * 64 + 2 * khalf;
        float ss = 0.0f;
#pragma unroll
        for (int kc = 0; kc < 16; ++kc) {
            v2f v = *(const v2f*)(xrow + 4 * kc);
            a[mt][kc] = v;
            ss += v.x * v.x + v.y * v.y;
        }
        ss += __shfl_xor(ss, 16, 32);   // lanes l / l+16 hold halves of the same row
        const float rn = rsqrtf(ss);
#pragma unroll
        for (int kc = 0; kc < 16; ++kc) { a[mt][kc].x *= rn; a[mt][kc].y *= rn; }
    }

    v8f acc[MROWS][4];
#pragma unroll
    for (int mt = 0; mt < MROWS; ++mt)
#pragma unroll
        for (int t = 0; t < 4; ++t) acc[mt][t] = {};

    for (int D = 0; D < DIM; D += 16) {
        const float* wrp = Wr + (size_t)(D + mrow) * HID + 2 * khalf;
        const float* wip = Wi + (size_t)(D + mrow) * HID + 2 * khalf;

        v8f pr[MROWS], pi[MROWS];
#pragma unroll
        for (int mt = 0; mt < MROWS; ++mt) { pr[mt] = {}; pi[mt] = {}; }

#pragma unroll
        for (int kc = 0; kc < 16; ++kc) {
            v2f b = *(const v2f*)(wrp + 4 * kc);
#pragma unroll
            for (int mt = 0; mt < MROWS; ++mt)
                pr[mt] = __builtin_amdgcn_wmma_f32_16x16x4_f32(false, a[mt][kc], false, b,
                                                               (short)0, pr[mt], false, false);
        }
#pragma unroll
        for (int kc = 0; kc < 16; ++kc) {
            v2f b = *(const v2f*)(wip + 4 * kc);
#pragma unroll
            for (int mt = 0; mt < MROWS; ++mt)
                pi[mt] = __builtin_amdgcn_wmma_f32_16x16x4_f32(false, a[mt][kc], false, b,
                                                               (short)0, pi[mt], false, false);
        }

        // probs = Pr^2 + Pi^2, then C-layout -> A-layout via wave-private LDS.
#pragma unroll
        for (int mt = 0; mt < MROWS; ++mt) {
            const v8f pb = pr[mt] * pr[mt] + pi[mt] * pi[mt];
#pragma unroll
            for (int r = 0; r < 8; ++r)
                lds[wave][mt][(r + 8 * khalf) * 17 + mrow] = pb[r];
        }

        v2f a2[MROWS][4];
#pragma unroll
        for (int mt = 0; mt < MROWS; ++mt)
#pragma unroll
            for (int k2 = 0; k2 < 4; ++k2) {
                a2[mt][k2].x = lds[wave][mt][mrow * 17 + 4 * k2 + 2 * khalf + 0];
                a2[mt][k2].y = lds[wave][mt][mrow * 17 + 4 * k2 + 2 * khalf + 1];
            }

        // out(16x64) += probs(16x16) @ St[D:D+16][:]  (4 col-tiles x 4 K-chunks),
        // sign B-fragments shared by both row tiles.
        const float* stp = St + (size_t)D * HID;
#pragma unroll
        for (int t = 0; t < 4; ++t) {
#pragma unroll
            for (int k2 = 0; k2 < 4; ++k2) {
                const int kb = 4 * k2 + 2 * khalf;
                v2f b;
                b.x = stp[(size_t)(kb + 0) * HID + t * 16 + mrow];
                b.y = stp[(size_t)(kb + 1) * HID + t * 16 + mrow];
#pragma unroll
                for (int mt = 0; mt < MROWS; ++mt)
                    acc[mt][t] = __builtin_amdgcn_wmma_f32_16x16x4_f32(false, a2[mt][k2], false, b,
                                                                      (short)0, acc[mt][t], false, false);
            }
        }
    }

    // --- Store MROWS x 16 x 64 output tiles ---
#pragma unroll
    for (int mt = 0; mt < MROWS; ++mt)
#pragma unroll
        for (int t = 0; t < 4; ++t)
#pragma unroll
            for (int r = 0; r < 8; ++r)
                Out[(size_t)(R + mt * 16 + r + 8 * khalf) * 64 + t * 16 + mrow] = acc[mt][t][r];
}

// ---------------------------------------------------------------------------
extern "C" void kernel_launch(void* const* d_in, const int* in_sizes, int n_in,
                              void* d_out, int out_size, void* d_ws, size_t ws_size,
                              hipStream_t stream) {
    const float* x   = (const float*)d_in[0];   // (8, 2048, 64)
    const float* rx0 = (const float*)d_in[1];   // (10,)
    const float* ry0 = (const float*)d_in[2];   // (10,)
    const float* ry1 = (const float*)d_in[3];   // (10,)

    float* Wr = (float*)d_ws;                   // 1024*64 floats
    float* Wi = Wr + DIM * HID;                 // 1024*64 floats
    float* St = Wi + DIM * HID;                 // 1024*64 floats (768 KB total)

    const int M = in_sizes[0] / HID;            // 16384 tokens

    build_unitary<<<HID, 256, 0, stream>>>(rx0, ry0, ry1, Wr, Wi, St);
    qsa_fused<<<M / (16 * MROWS * WAVES), 32 * WAVES, 0, stream>>>(x, Wr, Wi, St, (float*)d_out);
}